// QuantumDifficultyAdjuster_8048768712744
// MI455X (gfx1250) — compile-verified
//
#include <hip/hip_runtime.h>

// ---------------------------------------------------------------------------
// QuantumDifficultyAdjuster — closed-form CDNA5 kernel
//
// q_out[b,h] = prod_{q=0..2h} z_q(b),
//   z_q(b)   = cos(x[b,q])*cos(theta[3q+1]) - cos(theta[3q])*sin(x[b,q])*sin(theta[3q+1])
// out[b,k]   = sum_h q_out[b,h]*fc_w[k,h] + fc_b[k]
//
// Final 4->3 projection done with V_WMMA_F32_16X16X4_F32 (two tiles of 16
// rows per 32-row wave tile). Memory-bound: ~5.5 MB total traffic.
// ---------------------------------------------------------------------------

typedef __attribute__((ext_vector_type(2))) float v2f;
typedef __attribute__((ext_vector_type(8))) float v8f;

#define QDA_BATCH 131072

__global__ __launch_bounds__(256) void qda_kernel(
    const float* __restrict__ x,      // [BATCH, 8]
    const float* __restrict__ theta,  // [24]
    const float* __restrict__ fcw,    // [3, 4] row-major
    const float* __restrict__ fcb,    // [3]
    float* __restrict__ out)          // [BATCH, 3]
{
  const int lane    = threadIdx.x & 31;
  const int waveId  = (int)((blockIdx.x * blockDim.x + threadIdx.x) >> 5);
  const int nWaves  = (int)((gridDim.x * blockDim.x) >> 5);
  const int n       = lane & 15;           // WMMA column (N)
  const bool hiHalf = lane >= 16;

  // --- per-qubit coefficients from theta (uniform -> scalar loads) ---------
  float Aq[8], Bq[8];
#pragma unroll
  for (int q = 0; q < 8; ++q) {
    float t0 = theta[3 * q + 0];
    float t1 = theta[3 * q + 1];
    float s1, c1;
    __sincosf(t1, &s1, &c1);
    Aq[q] = c1;
    Bq[q] = __cosf(t0) * s1;
  }

  // --- B matrix (fc_w^T, 4x16, N padded with zeros) and C (bias) ----------
  // ISA layout: B VGPR j, lanes 0-15 -> K=j row; lanes 16-31 -> K=j+2 row.
  v2f bmat;
  v8f cmat;
  float b0v = 0.0f, b1v = 0.0f, biasv = 0.0f;
  if (n < 3) {
    b0v   = fcw[n * 4 + (hiHalf ? 2 : 0)];
    b1v   = fcw[n * 4 + (hiHalf ? 3 : 1)];
    biasv = fcb[n];
  }
  bmat.x = b0v;
  bmat.y = b1v;
#pragma unroll
  for (int j = 0; j < 8; ++j) cmat[j] = biasv;   // bias independent of M row

  // --- grid-stride over 32-row tiles (one row per lane) --------------------
  for (int tile = waveId; tile * 32 < QDA_BATCH; tile += nWaves) {
    const int rowBase = tile * 32;
    const int row     = rowBase + lane;

    const float4* xv = (const float4*)(x + (size_t)row * 8);
    float4 xa = xv[0];
    float4 xb = xv[1];

    float z[8];
    {
      float xs[8] = {xa.x, xa.y, xa.z, xa.w, xb.x, xb.y, xb.z, xb.w};
#pragma unroll
      for (int q = 0; q < 8; ++q) {
        float sv, cv;
        __sincosf(xs[q], &sv, &cv);
        z[q] = cv * Aq[q] - sv * Bq[q];
      }
    }

    // head products (qubit 7 never contributes)
    float p0 = z[0];
    float p1 = p0 * z[1] * z[2];
    float p2 = p1 * z[3] * z[4];
    float p3 = p2 * z[5] * z[6];

    // Repack into WMMA A-layout for two 16-row tiles.
    // Tile0 (rows rowBase..+15):  lanes 0-15 need (p0,p1) of row lane (own),
    //                             lanes 16-31 need (p2,p3) of row lane-16.
    // Tile1 (rows rowBase+16..31): lanes 0-15 need (p0,p1) of row lane+16,
    //                              lanes 16-31 need (p2,p3) of row lane (own).
    float q2lo = __shfl(p2, lane & 15, 32);
    float q3lo = __shfl(p3, lane & 15, 32);
    float q0hi = __shfl(p0, lane | 16, 32);
    float q1hi = __shfl(p1, lane | 16, 32);

    v2f a0, a1;
    a0.x = hiHalf ? q2lo : p0;
    a0.y = hiHalf ? q3lo : p1;
    a1.x = hiHalf ? p2 : q0hi;
    a1.y = hiHalf ? p3 : q1hi;

    // D = A*B + C  (8 args: neg_a, A, neg_b, B, c_mod, C, reuse_a, reuse_b)
    v8f d0 = __builtin_amdgcn_wmma_f32_16x16x4_f32(
        false, a0, false, bmat, (short)0, cmat, false, false);
    v8f d1 = __builtin_amdgcn_wmma_f32_16x16x4_f32(
        false, a1, false, bmat, (short)0, cmat, false, false);

    // Store: D VGPR j holds M = j + (hiHalf ? 8 : 0), N = lane & 15.
    if (n < 3) {
      const int mOff = hiHalf ? 8 : 0;
#pragma unroll
      for (int j = 0; j < 8; ++j) {
        out[(size_t)(rowBase + mOff + j) * 3 + n]      = d0[j];
        out[(size_t)(rowBase + 16 + mOff + j) * 3 + n] = d1[j];
      }
    }
  }
}

extern "C" void kernel_launch(void* const* d_in, const int* in_sizes, int n_in,
                              void* d_out, int out_size, void* d_ws, size_t ws_size,
                              hipStream_t stream) {
  const float* x     = (const float*)d_in[0];   // [131072, 8]
  const float* theta = (const float*)d_in[1];   // [24]
  const float* fcw   = (const float*)d_in[2];   // [3, 4]
  const float* fcb   = (const float*)d_in[3];   // [3]
  float*       out   = (float*)d_out;           // [131072, 3]

  (void)in_sizes; (void)n_in; (void)out_size; (void)d_ws; (void)ws_size;

  // 512 blocks * 8 waves = 4096 waves = BATCH/32 tiles, one tile per wave.
  qda_kernel<<<512, 256, 0, stream>>>(x, theta, fcw, fcb, out);
}